// SANet_9466107920434
// MI455X (gfx1250) — compile-verified
//
#include <hip/hip_runtime.h>
#include <hip/hip_bf16.h>

typedef __attribute__((ext_vector_type(2))) float v2f;
typedef __attribute__((ext_vector_type(8))) float v8f;
typedef int v4i __attribute__((vector_size(16)));

#define N_HEADC 8
#define D_HIDC  512
#define D_HEADC 64
#define D_MIDC  32
#define BATCHC  8
#define SEQC    128

// workspace layout (float offsets)
#define QH_OFF   0
#define KH_OFF   (BATCHC*N_HEADC*SEQC*D_HEADC)              // 524288
#define SM_OFF   (2*BATCHC*N_HEADC*SEQC*D_HEADC)            // 1048576
#define BIN_OFF  (SM_OFF + BATCHC*N_HEADC*SEQC*SEQC)        // 2097152

#define AS1 __attribute__((address_space(1)))
#define AS3 __attribute__((address_space(3)))

#if defined(__has_builtin)
#if __has_builtin(__builtin_amdgcn_global_load_async_to_lds_b128)
#define HAVE_ASYNC_LDS 1
#endif
#endif

__device__ __forceinline__ void wait_asynccnt0() {
#if defined(__has_builtin) && __has_builtin(__builtin_amdgcn_s_wait_asynccnt)
    __builtin_amdgcn_s_wait_asynccnt(0);
#else
    asm volatile("s_wait_asynccnt 0" ::: "memory");
#endif
}

// ---------------------------------------------------------------------------
// Kernel 1: q/k projections with fused bias, head-major output.
// One wave computes a 16x16 tile of X @ W^T using V_WMMA_F32_16X16X4_F32.
// ---------------------------------------------------------------------------
__global__ __launch_bounds__(128)
void proj_kernel(const float* __restrict__ q, const float* __restrict__ k,
                 const float* __restrict__ Wq, const float* __restrict__ bq,
                 const float* __restrict__ Wk, const float* __restrict__ bk,
                 float* __restrict__ ws)
{
    const int tid  = threadIdx.x;
    const int lane = tid & 31;
    const int wave = tid >> 5;
    const int gw   = blockIdx.x * 4 + wave;   // 0..4095
    const int which = gw >> 11;               // 0 = q-projection, 1 = k-projection
    const int t  = gw & 2047;                 // 64 M-tiles x 32 N-tiles
    const int m0 = (t >> 5) << 4;             // row base   (0..1008)
    const int n0 = (t & 31) << 4;             // col base   (0..496)

    const float* X    = which ? k  : q;
    const float* W    = which ? Wk : Wq;
    const float* bias = which ? bk : bq;
    float*       dst  = ws + (which ? KH_OFF : QH_OFF);

    const int half = lane >> 4;
    const int l16  = lane & 15;

    // A: 16x4 fp32 fragment — lane = row, half selects K / K+2
    const float* aptr = X + (m0 + l16) * D_HIDC + 2 * half;
    // B: 4x16 fp32 fragment — lane = col, half selects K / K+2
    const float* bptr = W + (n0 + l16) * D_HIDC + 2 * half;

    v8f acc = {0.f,0.f,0.f,0.f,0.f,0.f,0.f,0.f};
    for (int s = 0; s < D_HIDC / 4; ++s) {
        v2f a = *(const v2f*)(aptr + 4 * s);
        v2f b = *(const v2f*)(bptr + 4 * s);
        acc = __builtin_amdgcn_wmma_f32_16x16x4_f32(false, a, false, b,
                                                    (short)0, acc, false, false);
    }

    // bias + permuted store: row -> (batch, l), col -> (head, d)
    const float bv = bias[n0 + l16];
    const int h  = n0 >> 6;
    const int dd = (n0 & 63) + l16;
    #pragma unroll
    for (int r = 0; r < 8; ++r) {
        int row = m0 + r + 8 * half;
        int bb  = row >> 7;
        int l   = row & 127;
        dst[(((bb * N_HEADC + h) * SEQC) + l) * D_HEADC + dd] = acc[r] + bv;
    }
}

// ---------------------------------------------------------------------------
// Kernel 2: gate + score. One wave per (b,h,q).
// h[m,k] = ((W1ext ⊙ qh_q) @ kh^T)[m,k] with W1ext row 32 = ones giving the
// raw attention score in the same WMMA pass. A-fragments are register-cached.
// ---------------------------------------------------------------------------
__global__ __launch_bounds__(256)
void gate_kernel(const float* __restrict__ W1, const float* __restrict__ b1,
                 const float* __restrict__ W2, const float* __restrict__ b2,
                 float* __restrict__ ws)
{
    __shared__ float            w1e[48 * 64];     // 12 KB
    __shared__ __align__(16) float khs[128 * 64]; // 32 KB kh tile for this (b,h)
    __shared__ float            qrow[8 * 64];     //  2 KB
    __shared__ float            b1s[32];
    __shared__ float            w2s[64];

    const int tid  = threadIdx.x;
    const int lane = tid & 31;
    const int wave = tid >> 5;
    const int bx   = blockIdx.x;              // 0..1023 = (b, h, qgroup)
    const int b    = bx >> 7;
    const int h    = (bx >> 4) & 7;
    const int qg   = bx & 15;
    const int qidx = qg * 8 + wave;

    const float* qh = ws + QH_OFF + ((b * N_HEADC + h) * SEQC) * D_HEADC;
    const float* kh = ws + KH_OFF + ((b * N_HEADC + h) * SEQC) * D_HEADC;
    float* smp  = ws + SM_OFF  + ((b * N_HEADC + h) * SEQC + qidx) * SEQC;
    float* binp = ws + BIN_OFF + ((b * N_HEADC + h) * SEQC + qidx) * SEQC;

#ifdef HAVE_ASYNC_LDS
    // Stage the 32 KB kh tile via the async DMA path (ASYNCcnt-tracked),
    // overlapping it with the W1ext fill below. 16 B per lane per issue.
    #pragma unroll
    for (int j = 0; j < 8; ++j) {
        const int f4 = tid + 256 * j;             // float4 index, 2048 total
        AS1 v4i* gsrc =
            (AS1 v4i*)(unsigned long long)(kh + 4 * f4);
        AS3 v4i* ldst =
            (AS3 v4i*)(unsigned int)(unsigned long long)(&khs[4 * f4]);
        __builtin_amdgcn_global_load_async_to_lds_b128(gsrc, ldst, 0, 0);
    }
#else
    for (int i = tid; i < 128 * 64; i += 256) khs[i] = kh[i];
#endif

    for (int i = tid; i < 48 * 64; i += 256) {
        int m = i >> 6, d = i & 63;
        float v = 0.f;
        if (m < 32)       v = W1[m * 64 + d];
        else if (m == 32) v = 1.f;
        w1e[i] = v;
    }
    qrow[wave * 64 + lane]      = qh[qidx * 64 + lane];
    qrow[wave * 64 + lane + 32] = qh[qidx * 64 + lane + 32];
    if (tid < 32) b1s[tid] = b1[tid];
    if (tid < 64) w2s[tid] = W2[tid];
#ifdef HAVE_ASYNC_LDS
    wait_asynccnt0();
#endif
    __syncthreads();

    const float b20 = b2[0], b21 = b2[1];
    const int half = lane >> 4;
    const int l16  = lane & 15;
    const float* qr = qrow + wave * 64;

    // Register-cache all A fragments: a{0,1,2}[s] = W1ext-row-block ⊙ qh_q
    v2f a0[16], a1[16], a2[16];
    #pragma unroll
    for (int s = 0; s < 16; ++s) {
        const int kk = 4 * s + 2 * half;
        v2f qv = *(const v2f*)(qr + kk);
        a0[s] = *(const v2f*)(w1e + (l16     ) * 64 + kk) * qv;
        a1[s] = *(const v2f*)(w1e + (l16 + 16) * 64 + kk) * qv;
        a2[s] = *(const v2f*)(w1e + (l16 + 32) * 64 + kk) * qv;
    }

    for (int kt = 0; kt < 8; ++kt) {
        v8f d0 = {0,0,0,0,0,0,0,0};
        v8f d1 = {0,0,0,0,0,0,0,0};
        v8f d2 = {0,0,0,0,0,0,0,0};
        const float* kbase = khs + (kt * 16 + l16) * 64 + 2 * half;
        #pragma unroll
        for (int s = 0; s < 16; ++s) {
            v2f bf = *(const v2f*)(kbase + 4 * s);
            d0 = __builtin_amdgcn_wmma_f32_16x16x4_f32(false, a0[s], false, bf, (short)0, d0, false, false);
            d1 = __builtin_amdgcn_wmma_f32_16x16x4_f32(false, a1[s], false, bf, (short)0, d1, false, false);
            d2 = __builtin_amdgcn_wmma_f32_16x16x4_f32(false, a2[s], false, bf, (short)0, d2, false, false);
        }

        // Classifier: column k = kt*16 + l16 lives in lanes l16 and l16+16.
        // lanes<16 hold m = r and 16+r; lanes>=16 hold m = 8+r and 24+r.
        float c0 = 0.f, c1 = 0.f;
        const int mb = half * 8;
        #pragma unroll
        for (int r = 0; r < 8; ++r) {
            int m  = mb + r;
            float hv = fmaxf(d0[r] + b1s[m], 0.f);
            c0 += hv * w2s[m];
            c1 += hv * w2s[32 + m];
            int m2 = 16 + mb + r;
            float hv2 = fmaxf(d1[r] + b1s[m2], 0.f);
            c0 += hv2 * w2s[m2];
            c1 += hv2 * w2s[32 + m2];
        }
        c0 += __shfl_xor(c0, 16, 32);
        c1 += __shfl_xor(c1, 16, 32);
        c0 += b20; c1 += b21;
        float bin = (c1 > c0) ? 1.f : 0.f;   // argmax==1 (strict: ties pick class 0)

        float sraw = d2[0];                  // ones-row dot product, valid in lanes<16
        float sx   = __shfl_xor(sraw, 16, 32);
        float score = half ? sx : sraw;
        float smv = score * 0.125f * bin;    // /sqrt(64), masked

        if (half == 0) binp[kt * 16 + l16] = bin;
        else           smp [kt * 16 + l16] = smv;
    }
}

// ---------------------------------------------------------------------------
// Kernel 3: per-row L2 normalize + square, mean over heads. Block = (b,q).
// ---------------------------------------------------------------------------
__global__ __launch_bounds__(128)
void reduce_kernel(const float* __restrict__ ws, float* __restrict__ out)
{
    __shared__ float red[4];
    const int tid = threadIdx.x;      // k index
    const int bx  = blockIdx.x;       // b*128 + q
    const int b   = bx >> 7;
    const int q   = bx & 127;

    float accb = 0.f, acca = 0.f;
    for (int h = 0; h < 8; ++h) {
        const float* smp  = ws + SM_OFF  + ((b * 8 + h) * 128 + q) * 128;
        const float* binp = ws + BIN_OFF + ((b * 8 + h) * 128 + q) * 128;
        float s  = smp[tid];
        float bn = binp[tid];
        float ss = s * s;
        #pragma unroll
        for (int m = 16; m >= 1; m >>= 1) ss += __shfl_xor(ss, m, 32);
        if ((tid & 31) == 0) red[tid >> 5] = ss;
        __syncthreads();
        float tot = red[0] + red[1] + red[2] + red[3];
        __syncthreads();
        float nrm = fmaxf(sqrtf(tot), 1e-12f);
        float v = s / nrm;
        acca += v * v;
        accb += bn;
    }
    out[bx * 128 + tid]                        = accb * 0.125f;  // binary.mean(axis=1)
    out[BATCHC * SEQC * SEQC + bx * 128 + tid] = acca * 0.125f;  // attn.mean(axis=1)
}

// ---------------------------------------------------------------------------
extern "C" void kernel_launch(void* const* d_in, const int* in_sizes, int n_in,
                              void* d_out, int out_size, void* d_ws, size_t ws_size,
                              hipStream_t stream)
{
    const float* q  = (const float*)d_in[0];
    const float* k  = (const float*)d_in[1];
    const float* Wq = (const float*)d_in[2];
    const float* bq = (const float*)d_in[3];
    const float* Wk = (const float*)d_in[4];
    const float* bk = (const float*)d_in[5];
    const float* W1 = (const float*)d_in[6];
    const float* b1 = (const float*)d_in[7];
    const float* W2 = (const float*)d_in[8];
    const float* b2 = (const float*)d_in[9];
    float* ws  = (float*)d_ws;
    float* out = (float*)d_out;

    proj_kernel  <<<1024, 128, 0, stream>>>(q, k, Wq, bq, Wk, bk, ws);
    gate_kernel  <<<1024, 256, 0, stream>>>(W1, b1, W2, b2, ws);
    reduce_kernel<<<1024, 128, 0, stream>>>(ws, out);
}